// GraphVAE_63299228008753
// MI455X (gfx1250) — compile-verified
//
#include <hip/hip_runtime.h>

// ---------------------------------------------------------------------------
// GraphVAE forward for MI455X (gfx1250, wave32).
//   - All dense layers via V_WMMA_F32_16X16X4_F32 (full fp32 precision).
//   - Edge scatter via global f32 atomics (h1/agg fit in 192MB L2).
//   - BatchNorm: two-pass batch stats with LDS block reduction + few atomics.
// ---------------------------------------------------------------------------

typedef __attribute__((ext_vector_type(2))) float v2f;
typedef __attribute__((ext_vector_type(8))) float v8f;

#define ND 128   // input dim D
#define NH1 64
#define NH2 32
#define NL 128
#define NC 2
#define BN_EPS 1e-5f

// ---------------- WMMA GEMM:  C[M,NCOL] = A[M,K] @ B[K,NCOL] (+bias)(+relu)
// One wave computes a 16 x NCOL output tile. 4 waves / block.
// A-frag layout (16x4 f32): lanes 0-15 = M rows, half-lane selects K pair.
// C/D layout: vgpr r -> row (r + 8*halflane), lane&15 -> col within 16-group.
template <int K, int NCOL, bool RELU>
__global__ __launch_bounds__(128)
void wmma_gemm(const float* __restrict__ A, const float* __restrict__ B,
               const float* __restrict__ bias, float* __restrict__ C, int M) {
    constexpr int NT = NCOL / 16;
    const int wave = threadIdx.x >> 5;
    const int lane = threadIdx.x & 31;
    const int rowBase = (blockIdx.x * 4 + wave) * 16;
    if (rowBase >= M) return;            // uniform per wave -> EXEC stays all-1s

    const int hl = lane >> 4;            // 0: K pair {0,1}, 1: K pair {2,3}
    const int ll = lane & 15;

    v8f zero = {};
    v8f acc[NT];
#pragma unroll
    for (int n = 0; n < NT; ++n) acc[n] = zero;

    const float* __restrict__ arow = A + (size_t)(rowBase + ll) * K;

    for (int k0 = 0; k0 < K; k0 += 4) {
        v2f a;
        a.x = arow[k0 + hl * 2 + 0];
        a.y = arow[k0 + hl * 2 + 1];
#pragma unroll
        for (int n = 0; n < NT; ++n) {
            v2f b;
            b.x = B[(size_t)(k0 + hl * 2 + 0) * NCOL + n * 16 + ll];
            b.y = B[(size_t)(k0 + hl * 2 + 1) * NCOL + n * 16 + ll];
            acc[n] = __builtin_amdgcn_wmma_f32_16x16x4_f32(
                false, a, false, b, (short)0, acc[n], false, false);
        }
    }

#pragma unroll
    for (int n = 0; n < NT; ++n) {
        const int col = n * 16 + ll;
        const float bb = bias ? bias[col] : 0.0f;
#pragma unroll
        for (int r = 0; r < 8; ++r) {
            const int row = rowBase + r + hl * 8;
            float v = acc[n][r] + bb;
            if (RELU) v = fmaxf(v, 0.0f);
            C[(size_t)row * NCOL + col] = v;
        }
    }
}

// ---------------- degree ----------------------------------------------------
__global__ void deg_accum(const int* __restrict__ ei, float* __restrict__ deg,
                          long long nE) {
    long long e = (long long)blockIdx.x * blockDim.x + threadIdx.x;
    if (e >= nE) return;
    atomicAdd(&deg[ei[nE + e]], 1.0f);   // dst row of edge_index
}

__global__ void deg_finalize(float* __restrict__ dis, int n) {
    int i = blockIdx.x * blockDim.x + threadIdx.x;
    if (i < n) dis[i] = rsqrtf(dis[i] + 1.0f);
}

// ---------------- edge scatter: agg[dst] += h[src] * dis[src]*dis[dst] ------
// F/4 threads per edge, float4 per thread; stays L2-resident on MI455X.
template <int F>
__global__ void edge_scatter(const int* __restrict__ ei,
                             const float* __restrict__ dis,
                             const float* __restrict__ h,
                             float* __restrict__ agg, long long nE) {
    constexpr int TPE = F / 4;
    long long t = (long long)blockIdx.x * blockDim.x + threadIdx.x;
    long long e = t / TPE;
    int part = (int)(t % TPE);
    if (e >= nE) return;
    int s = ei[e];
    int d = ei[nE + e];
    float norm = dis[s] * dis[d];
    const float4 v = *(const float4*)(h + (size_t)s * F + part * 4);
    float* base = agg + (size_t)d * F + part * 4;
    atomicAdd(base + 0, v.x * norm);
    atomicAdd(base + 1, v.y * norm);
    atomicAdd(base + 2, v.z * norm);
    atomicAdd(base + 3, v.w * norm);
}

// ---------------- self-loop + bias, fused with BN batch-stat accumulation ---
// agg <- agg + h*dis^2 + bias ; stats[f] += sum, stats[F+f] += sumsq
template <int F>
__global__ __launch_bounds__(256)
void selfloop_bias_stats(const float* __restrict__ h,
                         const float* __restrict__ dis,
                         const float* __restrict__ bias,
                         float* __restrict__ agg,
                         float* __restrict__ stats, int n) {
    const int f = threadIdx.x % F;       // constant: 256 % F == 0, stride % F == 0
    const long long total = (long long)n * F;
    const long long stride = (long long)blockDim.x * gridDim.x;
    float s = 0.0f, ss = 0.0f;
    for (long long i = (long long)blockIdx.x * blockDim.x + threadIdx.x;
         i < total; i += stride) {
        const long long row = i / F;
        const float di = dis[row];
        const float v = agg[i] + h[i] * di * di + bias[f];
        agg[i] = v;
        s += v;
        ss += v * v;
    }
    __shared__ float ls[256], lss[256];
    ls[threadIdx.x] = s;
    lss[threadIdx.x] = ss;
    __syncthreads();
    if (threadIdx.x < F) {
        float a = 0.0f, b = 0.0f;
        for (int g = threadIdx.x; g < 256; g += F) { a += ls[g]; b += lss[g]; }
        atomicAdd(&stats[f], a);
        atomicAdd(&stats[F + f], b);
    }
}

__global__ void bn_finalize(float* __restrict__ stats, int F, float n) {
    int f = threadIdx.x;
    if (f >= F) return;
    float mean = stats[f] / n;
    float var = stats[F + f] / n - mean * mean;
    stats[2 * F + f] = mean;
    stats[3 * F + f] = rsqrtf(var + BN_EPS);
}

template <int F>
__global__ void bn_relu_apply(const float* __restrict__ pre,
                              const float* __restrict__ stats,
                              const float* __restrict__ gamma,
                              const float* __restrict__ beta,
                              float* __restrict__ out, int n) {
    long long i = (long long)blockIdx.x * blockDim.x + threadIdx.x;
    if (i >= (long long)n * F) return;
    int f = (int)(i % F);
    float v = (pre[i] - stats[2 * F + f]) * stats[3 * F + f] * gamma[f] + beta[f];
    out[i] = fmaxf(v, 0.0f);
}

// ---------------- pooling ---------------------------------------------------
__global__ void pool_accum(const float* __restrict__ h2,
                           const int* __restrict__ batch,
                           float* __restrict__ sums, float* __restrict__ cnt,
                           int n) {
    long long t = (long long)blockIdx.x * blockDim.x + threadIdx.x;
    long long i = t / NH2;
    int f = (int)(t % NH2);
    if (i >= n) return;
    int g = batch[i];
    atomicAdd(&sums[(size_t)g * NH2 + f], h2[(size_t)i * NH2 + f]);
    if (f == 0) atomicAdd(&cnt[g], 1.0f);
}

__global__ void pool_finalize(const float* __restrict__ sums,
                              const float* __restrict__ cnt,
                              float* __restrict__ pooled, int g_total) {
    int t = blockIdx.x * blockDim.x + threadIdx.x;
    if (t >= g_total * NH2) return;
    int g = t / NH2;
    pooled[t] = sums[t] / fmaxf(cnt[g], 1.0f);
}

// ---------------- reparameterize --------------------------------------------
__global__ void reparam(const float* __restrict__ mu,
                        const float* __restrict__ lv,
                        const float* __restrict__ eps,
                        float* __restrict__ z, int n) {
    int i = blockIdx.x * blockDim.x + threadIdx.x;
    if (i < n) z[i] = mu[i] + eps[i] * expf(0.5f * lv[i]);
}

// ---------------- classifier head (NCOL=2, not WMMA-shaped) -----------------
__global__ void classifier_head(const float* __restrict__ c1,
                                const float* __restrict__ Wc2,
                                const float* __restrict__ bc2,
                                float* __restrict__ out, int g_total) {
    int t = blockIdx.x * blockDim.x + threadIdx.x;
    if (t >= g_total * NC) return;
    int g = t / NC, c = t % NC;
    float acc = bc2[c];
    const float* row = c1 + (size_t)g * NH1;
#pragma unroll
    for (int k = 0; k < NH1; ++k) acc += row[k] * Wc2[k * NC + c];
    out[t] = acc;
}

// ---------------------------------------------------------------------------
extern "C" void kernel_launch(void* const* d_in, const int* in_sizes, int n_in,
                              void* d_out, int out_size, void* d_ws, size_t ws_size,
                              hipStream_t stream) {
    const float* x     = (const float*)d_in[0];
    const int*   ei    = (const int*)d_in[1];
    const int*   batch = (const int*)d_in[2];
    const float* W1 = (const float*)d_in[3],  *b1 = (const float*)d_in[4];
    const float* g1 = (const float*)d_in[5],  *be1 = (const float*)d_in[6];
    const float* W2 = (const float*)d_in[7],  *b2 = (const float*)d_in[8];
    const float* g2 = (const float*)d_in[9],  *be2 = (const float*)d_in[10];
    const float* Wmu = (const float*)d_in[11], *bmu = (const float*)d_in[12];
    const float* Wlv = (const float*)d_in[13], *blv = (const float*)d_in[14];
    const float* Wd1 = (const float*)d_in[15], *bd1 = (const float*)d_in[16];
    const float* Wd2 = (const float*)d_in[17], *bd2 = (const float*)d_in[18];
    const float* Wd3 = (const float*)d_in[19], *bd3 = (const float*)d_in[20];
    const float* Wc1 = (const float*)d_in[21], *bc1 = (const float*)d_in[22];
    const float* Wc2 = (const float*)d_in[23], *bc2 = (const float*)d_in[24];
    const float* eps = (const float*)d_in[25];

    const int N = in_sizes[0] / ND;
    const long long E = in_sizes[1] / 2;
    const int G = in_sizes[25] / NL;

    // ---- workspace layout (floats) ----
    float* ws = (float*)d_ws;
    size_t off = 0;
    float* dis    = ws + off; off += (size_t)N;
    float* h1lin  = ws + off; off += (size_t)N * NH1;   // reused as h1 post-BN
    float* agg1   = ws + off; off += (size_t)N * NH1;
    float* h2lin  = ws + off; off += (size_t)N * NH2;   // reused as h2 post-BN
    float* agg2   = ws + off; off += (size_t)N * NH2;
    float* stats1 = ws + off; off += 4 * NH1;
    float* stats2 = ws + off; off += 4 * NH2;
    float* psum   = ws + off; off += (size_t)G * NH2;
    float* pcnt   = ws + off; off += (size_t)G;
    float* pooled = ws + off; off += (size_t)G * NH2;
    float* dbuf1  = ws + off; off += (size_t)G * NH1;
    float* dbuf2  = ws + off; off += (size_t)G * NH1;
    float* cbuf   = ws + off; off += (size_t)G * NH1;

    // ---- output layout: classification, reconstructed, mu, logvar, z ----
    float* out        = (float*)d_out;
    float* out_cls    = out;
    float* out_recon  = out_cls   + (size_t)G * NC;
    float* out_mu     = out_recon + (size_t)G * ND;
    float* out_logvar = out_mu    + (size_t)G * NL;
    float* out_z      = out_logvar + (size_t)G * NL;

    // ---- zero accumulators ----
    hipMemsetAsync(dis,    0, (size_t)N * sizeof(float), stream);
    hipMemsetAsync(agg1,   0, (size_t)N * NH1 * sizeof(float), stream);
    hipMemsetAsync(agg2,   0, (size_t)N * NH2 * sizeof(float), stream);
    hipMemsetAsync(stats1, 0, 4 * NH1 * sizeof(float), stream);
    hipMemsetAsync(stats2, 0, 4 * NH2 * sizeof(float), stream);
    hipMemsetAsync(psum,   0, ((size_t)G * NH2 + G) * sizeof(float), stream);

    // ---- degree / normalization ----
    deg_accum<<<(unsigned)((E + 255) / 256), 256, 0, stream>>>(ei, dis, E);
    deg_finalize<<<(N + 255) / 256, 256, 0, stream>>>(dis, N);

    // ---- layer 1: GCNConv(D->H1) + BN + ReLU ----
    {
        unsigned blocks = (unsigned)((N / 16 + 3) / 4);
        wmma_gemm<ND, NH1, false><<<blocks, 128, 0, stream>>>(x, W1, nullptr, h1lin, N);
    }
    edge_scatter<NH1><<<(unsigned)((E * (NH1 / 4) + 255) / 256), 256, 0, stream>>>(
        ei, dis, h1lin, agg1, E);
    selfloop_bias_stats<NH1><<<512, 256, 0, stream>>>(h1lin, dis, b1, agg1, stats1, N);
    bn_finalize<<<1, 128, 0, stream>>>(stats1, NH1, (float)N);
    bn_relu_apply<NH1><<<(unsigned)(((size_t)N * NH1 + 255) / 256), 256, 0, stream>>>(
        agg1, stats1, g1, be1, h1lin, N);

    // ---- layer 2: GCNConv(H1->H2) + BN + ReLU ----
    {
        unsigned blocks = (unsigned)((N / 16 + 3) / 4);
        wmma_gemm<NH1, NH2, false><<<blocks, 128, 0, stream>>>(h1lin, W2, nullptr, h2lin, N);
    }
    edge_scatter<NH2><<<(unsigned)((E * (NH2 / 4) + 255) / 256), 256, 0, stream>>>(
        ei, dis, h2lin, agg2, E);
    selfloop_bias_stats<NH2><<<512, 256, 0, stream>>>(h2lin, dis, b2, agg2, stats2, N);
    bn_finalize<<<1, 64, 0, stream>>>(stats2, NH2, (float)N);
    bn_relu_apply<NH2><<<(unsigned)(((size_t)N * NH2 + 255) / 256), 256, 0, stream>>>(
        agg2, stats2, g2, be2, h2lin, N);

    // ---- global mean pool ----
    pool_accum<<<(unsigned)(((size_t)N * NH2 + 255) / 256), 256, 0, stream>>>(
        h2lin, batch, psum, pcnt, N);
    pool_finalize<<<(G * NH2 + 255) / 256, 256, 0, stream>>>(psum, pcnt, pooled, G);

    // ---- VAE heads ----
    unsigned gblocks = (unsigned)((G / 16 + 3) / 4);
    wmma_gemm<NH2, NL, false><<<gblocks, 128, 0, stream>>>(pooled, Wmu, bmu, out_mu, G);
    wmma_gemm<NH2, NL, false><<<gblocks, 128, 0, stream>>>(pooled, Wlv, blv, out_logvar, G);
    reparam<<<(G * NL + 255) / 256, 256, 0, stream>>>(out_mu, out_logvar, eps, out_z, G * NL);

    // ---- decoder ----
    wmma_gemm<NL, NH1, true ><<<gblocks, 128, 0, stream>>>(out_z, Wd1, bd1, dbuf1, G);
    wmma_gemm<NH1, NH1, true ><<<gblocks, 128, 0, stream>>>(dbuf1, Wd2, bd2, dbuf2, G);
    wmma_gemm<NH1, ND, false><<<gblocks, 128, 0, stream>>>(dbuf2, Wd3, bd3, out_recon, G);

    // ---- classifier ----
    wmma_gemm<NL, NH1, true ><<<gblocks, 128, 0, stream>>>(out_z, Wc1, bc1, cbuf, G);
    classifier_head<<<(G * NC + 255) / 256, 256, 0, stream>>>(cbuf, Wc2, bc2, out_cls, G);
}